// ACELoss_80504866996280
// MI455X (gfx1250) — compile-verified
//
#include <hip/hip_runtime.h>
#include <math.h>
#include <stdint.h>

#define NB 15
#define BLOCK 256
#define WPB (BLOCK / 32)
#define GRID 2048
#define MAXC4 250 /* supports C == 1000 (250 float4 per row) */

// ---- CDNA5 async global->LDS path (guarded so compile never hard-fails) ----
#if defined(__gfx1250__) && __has_builtin(__builtin_amdgcn_global_load_async_to_lds_b128)
#define HAVE_ASYNC 1
#else
#define HAVE_ASYNC 0
#endif

#if HAVE_ASYNC
#if __has_builtin(__builtin_amdgcn_s_wait_asynccnt)
#define WAIT_ASYNC(n) __builtin_amdgcn_s_wait_asynccnt(n)
#else
#define WAIT_ASYNC(n) asm volatile("s_wait_asynccnt " #n ::: "memory")
#endif
// Builtin signature (from hipcc diagnostic): param0 = v4i in addrspace(1),
// param1 = v4i in addrspace(3), then two immediate ints (offset, cpol).
typedef int v4i __attribute__((vector_size(16)));
#define AS1(p) ((__attribute__((address_space(1))) v4i*)(p))
#define AS3(p) ((__attribute__((address_space(3))) v4i*)(p))
#endif

// Bin boundaries: k/15 correctly rounded to float (matches float32 linspace to
// within the rounding of the reference itself).
__device__ __forceinline__ float bnd(int k) {
  const float b[16] = {0.0f,        1.0f / 15.0f,  2.0f / 15.0f,  3.0f / 15.0f,
                       4.0f / 15.0f, 5.0f / 15.0f, 6.0f / 15.0f,  7.0f / 15.0f,
                       8.0f / 15.0f, 9.0f / 15.0f, 10.0f / 15.0f, 11.0f / 15.0f,
                       12.0f / 15.0f, 13.0f / 15.0f, 14.0f / 15.0f, 1.0f};
  return b[k];
}

// idx = (#boundaries strictly below p) - 1, assuming p > 0.
__device__ __forceinline__ int bin_of(float p) {
  int b = 0;
#pragma unroll
  for (int j = 1; j <= 14; ++j) b += (p > bnd(j)) ? 1 : 0;
  return b;
}

__global__ __launch_bounds__(BLOCK) void ace_main(
    const float* __restrict__ logits, const int* __restrict__ labels, int rows,
    int C, float* __restrict__ g_conf, float* __restrict__ g_acc,
    unsigned* __restrict__ g_cnt) {
  __shared__ float s_conf[NB];
  __shared__ float s_acc[NB];
  __shared__ unsigned s_cnt[NB];
#if HAVE_ASYNC
  __shared__ float4 s_buf[WPB][2][MAXC4];
#endif
  const int tid = threadIdx.x;
  const int lane = tid & 31;
  const int w = tid >> 5;
  if (tid < NB) {
    s_conf[tid] = 0.0f;
    s_acc[tid] = 0.0f;
    s_cnt[tid] = 0u;
  }
  __syncthreads();

  const int C4 = C >> 2;
  const int gw = blockIdx.x * WPB + w;  // global wave id
  const int nw = gridDim.x * WPB;       // total waves

  unsigned cnt0 = 0u;
  float sum0 = 0.0f;
  const float B1 = bnd(1);
  const float NEG_INF = -__builtin_inff();

#if HAVE_ASYNC
  if (gw < rows) {
    const float4* src = (const float4*)(logits + (size_t)gw * C);
#pragma unroll
    for (int k = 0; k < 8; ++k) {
      int i4 = k * 32 + lane;
      if (i4 < C4)
        __builtin_amdgcn_global_load_async_to_lds_b128(
            AS1(src + i4), AS3(&s_buf[w][0][i4]), 0, 0);
    }
  }
  int cur = 0;
#endif

  for (int row = gw; row < rows; row += nw) {
    const int nrow = row + nw;
#if HAVE_ASYNC
    if (nrow < rows) {
      const float4* src = (const float4*)(logits + (size_t)nrow * C);
#pragma unroll
      for (int k = 0; k < 8; ++k) {
        int i4 = k * 32 + lane;
        if (i4 < C4)
          __builtin_amdgcn_global_load_async_to_lds_b128(
              AS1(src + i4), AS3(&s_buf[w][cur ^ 1][i4]), 0, 0);
      }
      WAIT_ASYNC(8);  // oldest 8 (current row) complete; next 8 in flight
    } else {
      WAIT_ASYNC(0);
    }
    const float4* rsrc = &s_buf[w][cur][0];
    cur ^= 1;
#else
    const float4* rsrc = (const float4*)(logits + (size_t)row * C);
    if (nrow < rows) {
      const char* np = (const char*)(logits + (size_t)nrow * C);
      __builtin_prefetch(np + lane * 128, 0, 1);  // global_prefetch_b8
    }
#endif
    // Load row slice into registers (ds_load_b128 from LDS on async path).
    float x[32];
#pragma unroll
    for (int k = 0; k < 8; ++k) {
      int i4 = k * 32 + lane;
      float4 v;
      if (i4 < C4) {
        v = rsrc[i4];
      } else {
        v.x = v.y = v.z = v.w = NEG_INF;
      }
      x[4 * k + 0] = v.x;
      x[4 * k + 1] = v.y;
      x[4 * k + 2] = v.z;
      x[4 * k + 3] = v.w;
    }
    // Row max (lane-local then wave32 allreduce).
    float mx = x[0];
#pragma unroll
    for (int i = 1; i < 32; ++i) mx = fmaxf(mx, x[i]);
#pragma unroll
    for (int off = 16; off > 0; off >>= 1)
      mx = fmaxf(mx, __shfl_xor(mx, off, 32));
    // exp + sum (overwrite x with exp values).
    float s = 0.0f;
#pragma unroll
    for (int i = 0; i < 32; ++i) {
      float e = __expf(x[i] - mx);
      x[i] = e;
      s += e;
    }
#pragma unroll
    for (int off = 16; off > 0; off >>= 1) s += __shfl_xor(s, off, 32);
    const float r = 1.0f / s;

    // Binning pass. Bin 0 (p <= 1/15) dominates with 1000 classes -> registers.
#pragma unroll
    for (int i = 0; i < 32; ++i) {
      float p = x[i] * r;
      if (p > 0.0f && p <= B1) {
        cnt0++;
        sum0 += p;
      } else if (p > B1 && p <= 1.0f) {  // rare: <= 14 per row
        int b = 1;
#pragma unroll
        for (int j = 2; j <= 14; ++j) b += (p > bnd(j)) ? 1 : 0;
        atomicAdd(&s_cnt[b], 1u);
        atomicAdd(&s_conf[b], p);
      }
    }
    // Accuracy contribution: one element (label class) per row.
    if (lane == 0) {
      int lbl = labels[row];
      float lg = logits[(size_t)row * C + lbl];
      float p = __expf(lg - mx) * r;
      if (p > 0.0f && p <= 1.0f) atomicAdd(&s_acc[bin_of(p)], 1.0f);
    }
  }

  // Wave-reduce the fast-bin accumulators, fold into block bins, then global.
#pragma unroll
  for (int off = 16; off > 0; off >>= 1) {
    sum0 += __shfl_xor(sum0, off, 32);
    cnt0 += __shfl_xor(cnt0, off, 32);
  }
  if (lane == 0 && cnt0 != 0u) {
    atomicAdd(&s_cnt[0], cnt0);
    atomicAdd(&s_conf[0], sum0);
  }
  __syncthreads();
  if (tid < NB) {
    if (s_cnt[tid] != 0u) atomicAdd(&g_cnt[tid], s_cnt[tid]);
    if (s_conf[tid] != 0.0f) atomicAdd(&g_conf[tid], s_conf[tid]);
    if (s_acc[tid] != 0.0f) atomicAdd(&g_acc[tid], s_acc[tid]);
  }
}

// ---------------- generic fallback (any C), block-per-row, 3 passes ---------
__device__ __forceinline__ float blockReduceMaxF(float v, float* scratch) {
#pragma unroll
  for (int off = 16; off > 0; off >>= 1) v = fmaxf(v, __shfl_xor(v, off, 32));
  const int lane = threadIdx.x & 31;
  const int w = threadIdx.x >> 5;
  __syncthreads();
  if (lane == 0) scratch[w] = v;
  __syncthreads();
  float r = scratch[0];
#pragma unroll
  for (int i = 1; i < WPB; ++i) r = fmaxf(r, scratch[i]);
  return r;
}

__device__ __forceinline__ float blockReduceSumF(float v, float* scratch) {
#pragma unroll
  for (int off = 16; off > 0; off >>= 1) v += __shfl_xor(v, off, 32);
  const int lane = threadIdx.x & 31;
  const int w = threadIdx.x >> 5;
  __syncthreads();
  if (lane == 0) scratch[w] = v;
  __syncthreads();
  float r = scratch[0];
#pragma unroll
  for (int i = 1; i < WPB; ++i) r += scratch[i];
  return r;
}

__global__ __launch_bounds__(BLOCK) void ace_generic(
    const float* __restrict__ logits, const int* __restrict__ labels, int rows,
    int C, float* __restrict__ g_conf, float* __restrict__ g_acc,
    unsigned* __restrict__ g_cnt) {
  __shared__ float s_conf[NB];
  __shared__ float s_acc[NB];
  __shared__ unsigned s_cnt[NB];
  __shared__ float s_red[WPB];
  const int tid = threadIdx.x;
  if (tid < NB) {
    s_conf[tid] = 0.0f;
    s_acc[tid] = 0.0f;
    s_cnt[tid] = 0u;
  }
  __syncthreads();
  for (int row = blockIdx.x; row < rows; row += gridDim.x) {
    const float* rp = logits + (size_t)row * C;
    float mx = -__builtin_inff();
    for (int j = tid; j < C; j += BLOCK) mx = fmaxf(mx, rp[j]);
    mx = blockReduceMaxF(mx, s_red);
    float s = 0.0f;
    for (int j = tid; j < C; j += BLOCK) s += __expf(rp[j] - mx);
    s = blockReduceSumF(s, s_red);
    const float r = 1.0f / s;
    const int lbl = labels[row];
    for (int j = tid; j < C; j += BLOCK) {
      float p = __expf(rp[j] - mx) * r;
      if (p > 0.0f && p <= 1.0f) {
        int b = bin_of(p);
        atomicAdd(&s_cnt[b], 1u);
        atomicAdd(&s_conf[b], p);
        if (j == lbl) atomicAdd(&s_acc[b], 1.0f);
      }
    }
  }
  __syncthreads();
  if (tid < NB) {
    if (s_cnt[tid] != 0u) atomicAdd(&g_cnt[tid], s_cnt[tid]);
    if (s_conf[tid] != 0.0f) atomicAdd(&g_conf[tid], s_conf[tid]);
    if (s_acc[tid] != 0.0f) atomicAdd(&g_acc[tid], s_acc[tid]);
  }
}

__global__ void ace_finalize(const float* __restrict__ g_conf,
                             const float* __restrict__ g_acc,
                             const unsigned* __restrict__ g_cnt,
                             float* __restrict__ out, float total) {
  if (threadIdx.x == 0 && blockIdx.x == 0) {
    float ece = 0.0f;
#pragma unroll
    for (int b = 0; b < NB; ++b) {
      unsigned c = g_cnt[b];
      if (c != 0u) {
        float n = (float)c;
        ece += fabsf(g_conf[b] / n - g_acc[b] / n) * (n / total);
      }
    }
    out[0] = ece;
  }
}

extern "C" void kernel_launch(void* const* d_in, const int* in_sizes, int n_in,
                              void* d_out, int out_size, void* d_ws,
                              size_t ws_size, hipStream_t stream) {
  const float* logits = (const float*)d_in[0];
  const int* labels = (const int*)d_in[1];
  const int rows = in_sizes[1];
  const int C = (rows > 0) ? (in_sizes[0] / rows) : 0;

  float* g_conf = (float*)d_ws;
  float* g_acc = g_conf + NB;
  unsigned* g_cnt = (unsigned*)(g_acc + NB);

  (void)hipMemsetAsync(d_ws, 0, 3 * NB * sizeof(float), stream);

  if (C > 0 && (C & 3) == 0 && (C >> 2) <= MAXC4) {
    ace_main<<<dim3(GRID), dim3(BLOCK), 0, stream>>>(logits, labels, rows, C,
                                                     g_conf, g_acc, g_cnt);
  } else if (C > 0) {
    int g = rows < 4096 ? (rows > 0 ? rows : 1) : 4096;
    ace_generic<<<dim3(g), dim3(BLOCK), 0, stream>>>(logits, labels, rows, C,
                                                     g_conf, g_acc, g_cnt);
  }
  const float total = (float)((double)rows * (double)C);
  ace_finalize<<<dim3(1), dim3(32), 0, stream>>>(g_conf, g_acc, g_cnt,
                                                 (float*)d_out, total);
}